// Retriever_34411277975611
// MI455X (gfx1250) — compile-verified
//
#include <hip/hip_runtime.h>
#include <hip/hip_bf16.h>

typedef float v2f __attribute__((ext_vector_type(2)));
typedef float v8f __attribute__((ext_vector_type(8)));

#define B_  128
#define E_  512
#define K_  5
#define L_  128
#define D_  128
#define N_  (B_ * E_)   // 65536

// ---------------------------------------------------------------------------
// Stage 1: per-row cosine similarity, 16 rows per wave via V_WMMA_F32_16X16X4_F32
//
// A (16x4 f32) layout: lanes 0-15 hold M=lane, VGPR0=K0/VGPR1=K1;
//                      lanes 16-31 hold M=lane-16, VGPR0=K2/VGPR1=K3.
// B (4x16 f32) mirrors it with N in place of M. Feeding claim rows as A and
// evidence rows as B makes diag(A*B^T)[m] = dot(claim[m], evidence[m]).
// The same per-lane data yields partial |a|^2,|b|^2; one xor-16 shuffle
// completes the norms.
// ---------------------------------------------------------------------------
__global__ __launch_bounds__(256) void cos_kernel(const float* __restrict__ claim,
                                                  const float* __restrict__ evid,
                                                  float* __restrict__ cosv) {
    const int lane    = threadIdx.x & 31;
    const int wave    = threadIdx.x >> 5;
    const int group   = blockIdx.x * 8 + wave;   // one wave = 16 row pairs
    const int rowBase = group * 16;
    const int m       = lane & 15;
    const int koff    = (lane >> 4) * 2;         // 0 for lanes 0-15, 2 for 16-31

    const float2* ap = reinterpret_cast<const float2*>(claim + (size_t)(rowBase + m) * D_ + koff);
    const float2* bp = reinterpret_cast<const float2*>(evid  + (size_t)(rowBase + m) * D_ + koff);

    v8f acc = {0.f, 0.f, 0.f, 0.f, 0.f, 0.f, 0.f, 0.f};
    float na2 = 0.f, nb2 = 0.f;

#pragma unroll
    for (int t = 0; t < D_ / 4; ++t) {           // 32 WMMA steps cover K=128
        float2 a = ap[2 * t];                    // row, cols 4t+koff .. +1
        float2 b = bp[2 * t];
        v2f av; av.x = a.x; av.y = a.y;
        v2f bv; bv.x = b.x; bv.y = b.y;
        acc = __builtin_amdgcn_wmma_f32_16x16x4_f32(
            /*neg_a=*/false, av, /*neg_b=*/false, bv,
            /*c_mod=*/(short)0, acc, /*reuse_a=*/false, /*reuse_b=*/false);
        na2 = fmaf(a.x, a.x, na2); na2 = fmaf(a.y, a.y, na2);
        nb2 = fmaf(b.x, b.x, nb2); nb2 = fmaf(b.y, b.y, nb2);
    }

    // complete row norms (lane m and lane m+16 each hold half of row m's sum)
    na2 += __shfl_xor(na2, 16, 32);
    nb2 += __shfl_xor(nb2, 16, 32);

    // diagonal extraction: C/D layout -> diag(M) lives at
    //   M<8 : VGPR M,   lane M
    //   M>=8: VGPR M-8, lane M+16
    // every lane exposes acc[lane&7]; destination lane m pulls from m or m+16.
    const int sel = lane & 7;
    float dv = acc[0];
    if (sel == 1) dv = acc[1];
    if (sel == 2) dv = acc[2];
    if (sel == 3) dv = acc[3];
    if (sel == 4) dv = acc[4];
    if (sel == 5) dv = acc[5];
    if (sel == 6) dv = acc[6];
    if (sel == 7) dv = acc[7];
    const int src = (m < 8) ? m : (m + 16);
    float dot = __shfl(dv, src, 32);

    if (lane < 16) {
        const float eps = 1e-8f;
        float den = fmaxf(sqrtf(na2), eps) * fmaxf(sqrtf(nb2), eps);
        cosv[rowBase + m] = dot / den;
    }
}

// ---------------------------------------------------------------------------
// Stage 2: per-batch softmax over E=512 + iterative top-5
// (softmax is monotone, so top-k of softmax == top-k of cos; tie-break: lower
// index first, matching jax.lax.top_k)
// ---------------------------------------------------------------------------
__global__ __launch_bounds__(256) void topk_kernel(const float* __restrict__ cosv,
                                                   float* __restrict__ top_cos,
                                                   int* __restrict__ top_idx) {
    __shared__ float sval[E_];
    __shared__ float red[256];
    __shared__ int   redi[256];

    const int b   = blockIdx.x;
    const int tid = threadIdx.x;

    float c0 = cosv[b * E_ + tid];
    float c1 = cosv[b * E_ + tid + 256];

    // row max
    red[tid] = fmaxf(c0, c1);
    __syncthreads();
    for (int s = 128; s > 0; s >>= 1) {
        if (tid < s) red[tid] = fmaxf(red[tid], red[tid + s]);
        __syncthreads();
    }
    const float rowmax = red[0];
    __syncthreads();

    // exp + sum
    float e0 = expf(c0 - rowmax);
    float e1 = expf(c1 - rowmax);
    red[tid] = e0 + e1;
    __syncthreads();
    for (int s = 128; s > 0; s >>= 1) {
        if (tid < s) red[tid] += red[tid + s];
        __syncthreads();
    }
    const float denom = red[0];
    __syncthreads();

    sval[tid]       = e0;
    sval[tid + 256] = e1;
    __syncthreads();

    for (int k = 0; k < K_; ++k) {
        float v0 = sval[tid], v1 = sval[tid + 256];
        float bv; int bi;
        if (v0 >= v1) { bv = v0; bi = tid; }         // tie -> lower index
        else          { bv = v1; bi = tid + 256; }
        red[tid] = bv; redi[tid] = bi;
        __syncthreads();
        for (int s = 128; s > 0; s >>= 1) {
            if (tid < s) {
                float ov = red[tid + s]; int oi = redi[tid + s];
                if (ov > red[tid] || (ov == red[tid] && oi < redi[tid])) {
                    red[tid] = ov; redi[tid] = oi;
                }
            }
            __syncthreads();
        }
        if (tid == 0) {
            int idx = redi[0];
            top_cos[b * K_ + k] = red[0] / denom;    // softmax probability
            top_idx[b * K_ + k] = idx;
            sval[idx] = -1.f;                        // exclude (softmax vals > 0)
        }
        __syncthreads();
    }
}

// ---------------------------------------------------------------------------
// Stage 3: gather 6 int arrays (claim: first K rows / evidence: top-k rows),
// converting to float for the flat float output buffer.
// Output layout: [0,640) top_cos | then 6 sections of 640*128:
//   ids_claim, ids_evidence, masks_claim, masks_evidence, token_claim, token_ev
// ---------------------------------------------------------------------------
__global__ __launch_bounds__(128) void gather_kernel(const int* __restrict__ ids_c,
                                                     const int* __restrict__ ids_e,
                                                     const int* __restrict__ m_c,
                                                     const int* __restrict__ m_e,
                                                     const int* __restrict__ t_c,
                                                     const int* __restrict__ t_e,
                                                     const int* __restrict__ top_idx,
                                                     float* __restrict__ out) {
    const int r = blockIdx.x % (B_ * K_);   // 0..639
    const int s = blockIdx.x / (B_ * K_);   // 0..5 section
    const int b = r / K_;
    const int k = r % K_;

    const int* src = (s == 0) ? ids_c
                   : (s == 1) ? ids_e
                   : (s == 2) ? m_c
                   : (s == 3) ? m_e
                   : (s == 4) ? t_c
                   :            t_e;

    const int srcRow = (s & 1) ? (b * E_ + top_idx[b * K_ + k])  // evidence: top-k
                               : (b * E_ + k);                   // claim: first K

    float v = (float)src[(size_t)srcRow * L_ + threadIdx.x];
    out[(size_t)(B_ * K_) + (size_t)s * (B_ * K_ * L_) + (size_t)r * L_ + threadIdx.x] = v;
}

extern "C" void kernel_launch(void* const* d_in, const int* in_sizes, int n_in,
                              void* d_out, int out_size, void* d_ws, size_t ws_size,
                              hipStream_t stream) {
    (void)in_sizes; (void)n_in; (void)out_size; (void)ws_size;

    const float* claim = (const float*)d_in[0];
    const float* evid  = (const float*)d_in[1];
    const int*   ids_c = (const int*)d_in[2];
    const int*   ids_e = (const int*)d_in[3];
    const int*   m_c   = (const int*)d_in[4];
    const int*   m_e   = (const int*)d_in[5];
    const int*   t_c   = (const int*)d_in[6];
    const int*   t_e   = (const int*)d_in[7];

    float* out     = (float*)d_out;
    float* cosv    = (float*)d_ws;                               // N_ floats
    int*   top_idx = (int*)((char*)d_ws + (size_t)N_ * sizeof(float)); // 640 ints

    // Stage 1: 512 blocks x 256 threads; each wave handles 16 row pairs.
    cos_kernel<<<N_ / 128, 256, 0, stream>>>(claim, evid, cosv);

    // Stage 2: one block per batch row.
    topk_kernel<<<B_, 256, 0, stream>>>(cosv, out, top_idx);

    // Stage 3: 6 sections x 640 rows, one block per output row.
    gather_kernel<<<6 * B_ * K_, 128, 0, stream>>>(ids_c, ids_e, m_c, m_e,
                                                   t_c, t_e, top_idx, out);
}